// CustomLoss_52115133170353
// MI455X (gfx1250) — compile-verified
//
#include <hip/hip_runtime.h>
#include <math.h>

// CustomLoss: loss = mean_i( logsumexp(out[i,:]) - out[i, t_i] );
//             result = loss * (1 + mean(t_i in {3,5,8,9}))
// B = 262144, C = 1000, fp32. Memory-bound: 1.05 GB single pass -> ~45us @ 23.3TB/s.
//
// Kernel 1: one wave32 per row, row held in 8 float4 registers (NT b128 loads),
//           exact logsumexp, wave shuffle reduce, one float2 partial / block.
//           Specialized at compile time for C = 1000 (7 unconditional chunks +
//           1 lane-masked tail); generic fallback for other C.
// Kernel 2: single block folds the 32768 partials deterministically (no atomics).

typedef float v4f __attribute__((ext_vector_type(4)));

#define WAVES_PER_BLOCK 8
#define BLOCK_THREADS   (WAVES_PER_BLOCK * 32)

__device__ __forceinline__ float wave_max(float m) {
    #pragma unroll
    for (int off = 16; off > 0; off >>= 1)
        m = fmaxf(m, __shfl_xor(m, off, 32));
    return m;
}
__device__ __forceinline__ float wave_sum(float s) {
    #pragma unroll
    for (int off = 16; off > 0; off >>= 1)
        s += __shfl_xor(s, off, 32);
    return s;
}

// ---- Specialized: NVEC4 known at compile time, 32*8 >= NVEC4 > 32*7 --------
template <int NVEC4>
__global__ __launch_bounds__(BLOCK_THREADS)
void ce_loss_rows_fixed(const float* __restrict__ out,
                        const int*  __restrict__ tgt,
                        float2*     __restrict__ partials,
                        int B)
{
    __shared__ float2 wave_res[WAVES_PER_BLOCK];

    const int lane = threadIdx.x & 31;
    const int wave = threadIdx.x >> 5;
    const long long row = (long long)blockIdx.x * WAVES_PER_BLOCK + wave;

    float2 res; res.x = 0.0f; res.y = 0.0f;

    if (row < (long long)B) {
        const float* __restrict__ rp  = out + row * (long long)(NVEC4 * 4);
        const v4f*   __restrict__ rp4 = (const v4f*)rp;

        // Wave-uniform gathers (same address across lanes -> one request, no
        // exec-mask branch). Issued early to overlap the row stream.
        const int t = tgt[row];

        // Stream the whole row into registers: lane handles vec4 idx = lane + 32k.
        v4f v[8];
        #pragma unroll
        for (int k = 0; k < 7; ++k)
            v[k] = __builtin_nontemporal_load(rp4 + (lane + 32 * k));   // TH=NT
        if (lane < NVEC4 - 224)
            v[7] = __builtin_nontemporal_load(rp4 + (lane + 224));
        else
            v[7] = (v4f){-INFINITY, -INFINITY, -INFINITY, -INFINITY};

        const float xt = rp[t];   // row is hot in L2 regardless of NT near-hint

        // Phase 1: row max (pads are -inf, harmless).
        float m = -INFINITY;
        #pragma unroll
        for (int k = 0; k < 8; ++k)
            m = fmaxf(m, fmaxf(fmaxf(v[k].x, v[k].y), fmaxf(v[k].z, v[k].w)));
        m = wave_max(m);

        // Phase 2: sum of exp(x - m); exp(-inf) = 0 kills the pads for free.
        float s = 0.0f;
        #pragma unroll
        for (int k = 0; k < 8; ++k)
            s += __expf(v[k].x - m) + __expf(v[k].y - m)
               + __expf(v[k].z - m) + __expf(v[k].w - m);
        s = wave_sum(s);

        const float nll = (m + __logf(s)) - xt;
        const float msk = (t == 3 || t == 5 || t == 8 || t == 9) ? 1.0f : 0.0f;
        res.x = nll; res.y = msk;           // identical on all lanes; lane 0 writes
    }

    if (lane == 0) wave_res[wave] = res;
    __syncthreads();

    if (threadIdx.x == 0) {
        float a = 0.0f, b = 0.0f;
        #pragma unroll
        for (int w = 0; w < WAVES_PER_BLOCK; ++w) { a += wave_res[w].x; b += wave_res[w].y; }
        float2 p; p.x = a; p.y = b;
        partials[blockIdx.x] = p;
    }
}

// ---- Generic fallback (any C % 4 == 0), two passes over L2-resident row ----
__global__ __launch_bounds__(BLOCK_THREADS)
void ce_loss_rows_generic(const float* __restrict__ out,
                          const int*  __restrict__ tgt,
                          float2*     __restrict__ partials,
                          int B, int C, int nvec4)
{
    __shared__ float2 wave_res[WAVES_PER_BLOCK];

    const int lane = threadIdx.x & 31;
    const int wave = threadIdx.x >> 5;
    const long long row = (long long)blockIdx.x * WAVES_PER_BLOCK + wave;

    float2 res; res.x = 0.0f; res.y = 0.0f;

    if (row < (long long)B) {
        const float* __restrict__ rp  = out + row * (long long)C;
        const v4f*   __restrict__ rp4 = (const v4f*)rp;
        const int t = tgt[row];

        float m = -INFINITY;
        for (int i = lane; i < nvec4; i += 32) {
            v4f x = rp4[i];
            m = fmaxf(m, fmaxf(fmaxf(x.x, x.y), fmaxf(x.z, x.w)));
        }
        m = wave_max(m);

        float s = 0.0f;
        for (int i = lane; i < nvec4; i += 32) {
            v4f x = rp4[i];
            s += __expf(x.x - m) + __expf(x.y - m)
               + __expf(x.z - m) + __expf(x.w - m);
        }
        s = wave_sum(s);

        const float xt  = rp[t];
        const float nll = (m + __logf(s)) - xt;
        const float msk = (t == 3 || t == 5 || t == 8 || t == 9) ? 1.0f : 0.0f;
        res.x = nll; res.y = msk;
    }

    if (lane == 0) wave_res[wave] = res;
    __syncthreads();

    if (threadIdx.x == 0) {
        float a = 0.0f, b = 0.0f;
        #pragma unroll
        for (int w = 0; w < WAVES_PER_BLOCK; ++w) { a += wave_res[w].x; b += wave_res[w].y; }
        float2 p; p.x = a; p.y = b;
        partials[blockIdx.x] = p;
    }
}

#define FINAL_THREADS 1024

__global__ __launch_bounds__(FINAL_THREADS)
void ce_loss_final(const float2* __restrict__ partials, int n,
                   float* __restrict__ out, float invB)
{
    __shared__ float sa[FINAL_THREADS];
    __shared__ float sb[FINAL_THREADS];

    float a = 0.0f, b = 0.0f;
    for (int i = threadIdx.x; i < n; i += FINAL_THREADS) {
        float2 p = partials[i];
        a += p.x; b += p.y;
    }
    sa[threadIdx.x] = a; sb[threadIdx.x] = b;
    __syncthreads();

    for (int off = FINAL_THREADS / 2; off > 0; off >>= 1) {
        if (threadIdx.x < off) {
            sa[threadIdx.x] += sa[threadIdx.x + off];
            sb[threadIdx.x] += sb[threadIdx.x + off];
        }
        __syncthreads();
    }

    if (threadIdx.x == 0) {
        const float loss = sa[0] * invB;   // mean nll
        const float frac = sb[0] * invB;   // mean mask
        out[0] = loss * (1.0f + frac);     // loss + loss*mask.mean()
    }
}

extern "C" void kernel_launch(void* const* d_in, const int* in_sizes, int n_in,
                              void* d_out, int out_size, void* d_ws, size_t ws_size,
                              hipStream_t stream)
{
    const float* output = (const float*)d_in[0];   // [B, C] fp32
    const int*   target = (const int*)d_in[1];     // [B]    (integer -> const int*)

    const int B = in_sizes[1];                     // 262144
    const int C = in_sizes[0] / B;                 // 1000
    const int nvec4 = C / 4;                       // 250

    const int blocks = (B + WAVES_PER_BLOCK - 1) / WAVES_PER_BLOCK;  // 32768
    float2* partials = (float2*)d_ws;              // 32768 * 8 B = 256 KB scratch

    if (C == 1000) {
        ce_loss_rows_fixed<250><<<blocks, BLOCK_THREADS, 0, stream>>>(
            output, target, partials, B);
    } else {
        ce_loss_rows_generic<<<blocks, BLOCK_THREADS, 0, stream>>>(
            output, target, partials, B, C, nvec4);
    }
    ce_loss_final<<<1, FINAL_THREADS, 0, stream>>>(partials, blocks,
                                                   (float*)d_out, 1.0f / (float)B);
}